// MeasurementLayer_35150012350919
// MI455X (gfx1250) — compile-verified
//
#include <hip/hip_runtime.h>

typedef float v2f __attribute__((ext_vector_type(2)));
typedef float v4f __attribute__((ext_vector_type(4)));
typedef float v8f __attribute__((ext_vector_type(8)));

#define NQ      14
#define NSTATES 16384   // 2^14
#define TPB     256
#define EPT     64      // elements per thread (one contiguous chunk)

// One block per batch row. n = tid*64 + i*4 + e
//   bits 0-1 (e)   -> qubits 13,12 : in-float4 butterfly
//   bits 2-5 (i)   -> qubits 11-8  : compile-time +/- of per-iter chunk sum
//   bits 6-13(tid) -> qubits 7-0   : per-thread constant sign on thread total
// Final 256x16 -> 16 reduction via v_wmma_f32_16x16x4_f32 with A = ones.
__global__ __launch_bounds__(TPB)
void measurez_kernel(const float* __restrict__ sr,
                     const float* __restrict__ si,
                     float* __restrict__ out) {
    __shared__ float lds[TPB * 16];   // per-thread 16-vectors (16 KB)
    __shared__ float wsum[8 * 16];    // per-wave column partials

    const int tid = threadIdx.x;
    const int b   = blockIdx.x;

    const v4f* r4 = (const v4f*)(sr + (size_t)b * NSTATES + tid * EPT);
    const v4f* m4 = (const v4f*)(si + (size_t)b * NSTATES + tid * EPT);

    float T = 0.f;                    // thread total
    float aE0 = 0.f, aE1 = 0.f;       // e-bit signed sums (qubits 13,12)
    float aI0 = 0.f, aI1 = 0.f, aI2 = 0.f, aI3 = 0.f;  // i-bit sums (q11..q8)

#pragma unroll
    for (int i = 0; i < 16; ++i) {
        v4f r = __builtin_nontemporal_load(&r4[i]);   // global_load_b128, NT (streaming)
        v4f m = __builtin_nontemporal_load(&m4[i]);
        float p0 = fmaf(m.x, m.x, r.x * r.x);
        float p1 = fmaf(m.y, m.y, r.y * r.y);
        float p2 = fmaf(m.z, m.z, r.z * r.z);
        float p3 = fmaf(m.w, m.w, r.w * r.w);
        float s01 = p0 + p1, s23 = p2 + p3;
        float d01 = p0 - p1, d23 = p2 - p3;
        float S  = s01 + s23;         // chunk-of-4 total
        aE0 += d01 + d23;             // sign by bit0 of n
        aE1 += s01 - s23;             // sign by bit1 of n
        T   += S;
        aI0 += (i & 1) ? -S : S;      // folded to v_add/v_sub by unroll
        aI1 += (i & 2) ? -S : S;
        aI2 += (i & 4) ? -S : S;
        aI3 += (i & 8) ? -S : S;
    }

    float vals[16];
    vals[13] = aE0;  vals[12] = aE1;
    vals[11] = aI0;  vals[10] = aI1;  vals[9] = aI2;  vals[8] = aI3;
#pragma unroll
    for (int k = 0; k < 8; ++k)       // n bit (6+k) -> qubit 7-k, per-thread const sign
        vals[7 - k] = ((tid >> k) & 1) ? -T : T;
    vals[14] = 0.f;  vals[15] = 0.f;

#pragma unroll
    for (int j = 0; j < 4; ++j) {
        v4f t;
        t.x = vals[4 * j + 0];  t.y = vals[4 * j + 1];
        t.z = vals[4 * j + 2];  t.w = vals[4 * j + 3];
        *(v4f*)&lds[tid * 16 + 4 * j] = t;            // ds_store_b128
    }
    __syncthreads();

    // Each wave sums its 32 rows of lds via 8 accumulated WMMAs (A = ones(16x4)).
    const int wave = tid >> 5;
    const int lane = tid & 31;
    const int col  = lane & 15;           // B/D column = lane % 16 (ISA layout)
    const int krow = (lane >> 4) << 1;    // lanes 0-15: K rows 0,1; lanes 16-31: 2,3

    v2f ones; ones.x = 1.0f; ones.y = 1.0f;
    v8f acc = {};
#pragma unroll
    for (int t = 0; t < 8; ++t) {
        int row = (wave << 5) + (t << 2) + krow;
        v2f bm;
        bm.x = lds[row * 16 + col];
        bm.y = lds[(row + 1) * 16 + col];
        // D[m,q] = sum_k B[k,q] + C[m,q]  (A all-ones): column reduction
        acc = __builtin_amdgcn_wmma_f32_16x16x4_f32(
            /*neg_a=*/false, ones, /*neg_b=*/false, bm,
            /*c_mod=*/(short)0, acc, /*reuse_a=*/false, /*reuse_b=*/false);
    }
    if (lane < 16)                        // D VGPR0, lanes 0-15: M=0, N=lane
        wsum[wave * 16 + lane] = acc[0];
    __syncthreads();

    if (tid < NQ) {
        float s = 0.f;
#pragma unroll
        for (int w = 0; w < 8; ++w) s += wsum[w * 16 + tid];
        out[(size_t)b * NQ + tid] = s;
    }
}

extern "C" void kernel_launch(void* const* d_in, const int* in_sizes, int n_in,
                              void* d_out, int out_size, void* d_ws, size_t ws_size,
                              hipStream_t stream) {
    (void)n_in; (void)d_ws; (void)ws_size; (void)out_size;
    const float* sr = (const float*)d_in[0];
    const float* si = (const float*)d_in[1];
    float* out = (float*)d_out;
    const int batch = in_sizes[0] / NSTATES;   // 4096
    measurez_kernel<<<dim3(batch), dim3(TPB), 0, stream>>>(sr, si, out);
}